// SIFSGR_36696200577629
// MI455X (gfx1250) — compile-verified
//
#include <hip/hip_runtime.h>
#include <math.h>

// ---------------- problem constants (match reference) ----------------
#define T_        8
#define N_SUB     20000
#define M_SUB     5000
#define H_        128
#define L_        2
#define NNZ_G_    200000
#define NNZ_E_    100000
#define ITEM_NUM_ 40000
#define USER_NUM_ 10000
#define USER_TOT_ (USER_NUM_ + T_ * M_SUB)   // 50000
#define ITEM_TOT_ (ITEM_NUM_ + T_ * N_SUB)   // 200000
#define LDS_PITCH 132                        // 128 + 4 pad -> conflict-free strided reads

typedef __attribute__((ext_vector_type(2))) float v2f;
typedef __attribute__((ext_vector_type(8))) float v8f;

// fast sigmoid: v_exp_f32 + v_rcp_f32 (1-ulp rcp instead of full IEEE divide)
__device__ __forceinline__ float sigmoidf_(float v) {
    return __builtin_amdgcn_rcpf(1.0f + __expf(-v));
}

// ---------------------------------------------------------------------
// K0: fill output: user static copy, user dynamic zero (scatter target),
//     item static copy. Item-dynamic region is fully overwritten later.
// ---------------------------------------------------------------------
__global__ void k_init(const float* __restrict__ user_base,
                       const float* __restrict__ item_base,
                       float* __restrict__ out) {
    long i = (long)blockIdx.x * blockDim.x + threadIdx.x;
    const long userStatic = (long)USER_NUM_ * H_;
    const long userAll    = (long)USER_TOT_ * H_;
    const long total      = userAll + (long)ITEM_NUM_ * H_;
    if (i >= total) return;
    if (i < userStatic)      out[i] = user_base[i];
    else if (i < userAll)    out[i] = 0.0f;
    else                     out[i] = item_base[i - userAll];
}

// ---------------------------------------------------------------------
// K1: gated fusion.  Per wave: one 16-node tile.
//   a = item_base[rev_i], d = dy_table[rev_latest_i]  (staged in LDS)
//   s0 = sigmoid(a W + b), s1 = sigmoid(d W + b)  via f32 WMMA
//   x = s0*a + s1*d  -> item-dynamic region of d_out
// ---------------------------------------------------------------------
__global__ void __launch_bounds__(64) k_gate(
    const float* __restrict__ item_base,
    const float* __restrict__ dy_table,
    const float* __restrict__ W_gate,   // [T][H][H]
    const float* __restrict__ b_gate,   // [T][H]
    const int*  __restrict__ rev_i,     // [T][N]
    const int*  __restrict__ rev_dy,    // [T][N]
    float* __restrict__ xout)           // [T*N][H]
{
    __shared__ float lds[2][2][16 * LDS_PITCH];
    const int wave = threadIdx.x >> 5;
    const int lane = threadIdx.x & 31;
    const int half = lane >> 4;        // which 16-lane half
    const int n16  = lane & 15;
    const int t    = blockIdx.y;
    const int tile = blockIdx.x * 2 + wave;   // 0..1249, exact (625*2)
    const int row0 = tile * 16;

    float* la = lds[wave][0];
    float* ld = lds[wave][1];

    // gather-stage both 16x128 tiles (one row of float4 per iteration)
    for (int r = 0; r < 16; ++r) {
        const int ia = rev_i [t * N_SUB + row0 + r];
        const int id = rev_dy[t * N_SUB + row0 + r];
        float4 va = *(const float4*)(item_base + (long)ia * H_ + lane * 4);
        float4 vd = *(const float4*)(dy_table  + (long)id * H_ + lane * 4);
        *(float4*)(la + r * LDS_PITCH + lane * 4) = va;
        *(float4*)(ld + r * LDS_PITCH + lane * 4) = vd;
    }
    __syncthreads();

    const float* Wg = W_gate + t * H_ * H_;
    for (int ct = 0; ct < 8; ++ct) {            // 8 column tiles of 16
        const float bias = b_gate[t * H_ + ct * 16 + n16];
        v8f acc0, acc1;
        #pragma unroll
        for (int r = 0; r < 8; ++r) { acc0[r] = bias; acc1[r] = bias; }

        #pragma unroll 4
        for (int kb = 0; kb < H_; kb += 4) {
            const int krow = kb + 2 * half;
            v2f af, df, bf;
            // A fragment: lane holds row M=n16, K = krow, krow+1
            af[0] = la[n16 * LDS_PITCH + krow];
            af[1] = la[n16 * LDS_PITCH + krow + 1];
            df[0] = ld[n16 * LDS_PITCH + krow];
            df[1] = ld[n16 * LDS_PITCH + krow + 1];
            // B fragment: lane holds col N=n16, rows krow, krow+1
            bf[0] = Wg[(krow    ) * H_ + ct * 16 + n16];
            bf[1] = Wg[(krow + 1) * H_ + ct * 16 + n16];
            acc0 = __builtin_amdgcn_wmma_f32_16x16x4_f32(false, af, false, bf,
                                                         (short)0, acc0, false, false);
            acc1 = __builtin_amdgcn_wmma_f32_16x16x4_f32(false, df, false, bf,
                                                         (short)0, acc1, false, false);
        }

        #pragma unroll
        for (int r = 0; r < 8; ++r) {
            const int m = r + 8 * half;         // C/D layout row
            const float s0 = sigmoidf_(acc0[r]);
            const float s1 = sigmoidf_(acc1[r]);
            const float av = la[m * LDS_PITCH + ct * 16 + n16];
            const float dv = ld[m * LDS_PITCH + ct * 16 + n16];
            xout[((long)t * N_SUB + row0 + m) * H_ + ct * 16 + n16] = s0 * av + s1 * dv;
        }
    }
}

// ---------------------------------------------------------------------
// K2: y = act(x) @ W_hgnn[t,l] + b_hgnn[t,l]   (act = relu if flag)
// ---------------------------------------------------------------------
__global__ void __launch_bounds__(64) k_dense(
    const float* __restrict__ xin,   // [T*N][H]
    const float* __restrict__ W,     // [T][L][H][H]
    const float* __restrict__ bvec,  // [T][L][H]
    float* __restrict__ yout,        // [T*N][H]
    int layer, int apply_relu)
{
    __shared__ float lds[2][16 * LDS_PITCH];
    const int wave = threadIdx.x >> 5;
    const int lane = threadIdx.x & 31;
    const int half = lane >> 4;
    const int n16  = lane & 15;
    const int t    = blockIdx.y;
    const int tile = blockIdx.x * 2 + wave;
    const int row0 = tile * 16;

    float* lx = lds[wave];
    for (int r = 0; r < 16; ++r) {
        float4 v = *(const float4*)(xin + ((long)t * N_SUB + row0 + r) * H_ + lane * 4);
        if (apply_relu) {
            v.x = fmaxf(v.x, 0.0f); v.y = fmaxf(v.y, 0.0f);
            v.z = fmaxf(v.z, 0.0f); v.w = fmaxf(v.w, 0.0f);
        }
        *(float4*)(lx + r * LDS_PITCH + lane * 4) = v;
    }
    __syncthreads();

    const float* Wl = W    + ((long)t * L_ + layer) * H_ * H_;
    const float* bl = bvec + ((long)t * L_ + layer) * H_;
    for (int ct = 0; ct < 8; ++ct) {
        const float bias = bl[ct * 16 + n16];
        v8f acc;
        #pragma unroll
        for (int r = 0; r < 8; ++r) acc[r] = bias;

        #pragma unroll 4
        for (int kb = 0; kb < H_; kb += 4) {
            const int krow = kb + 2 * half;
            v2f af, bf;
            af[0] = lx[n16 * LDS_PITCH + krow];
            af[1] = lx[n16 * LDS_PITCH + krow + 1];
            bf[0] = Wl[(krow    ) * H_ + ct * 16 + n16];
            bf[1] = Wl[(krow + 1) * H_ + ct * 16 + n16];
            acc = __builtin_amdgcn_wmma_f32_16x16x4_f32(false, af, false, bf,
                                                        (short)0, acc, false, false);
        }
        #pragma unroll
        for (int r = 0; r < 8; ++r) {
            const int m = r + 8 * half;
            yout[((long)t * N_SUB + row0 + m) * H_ + ct * 16 + n16] = acc[r];
        }
    }
}

// ---------------------------------------------------------------------
// K3: zero a buffer (scatter target init)
// ---------------------------------------------------------------------
__global__ void k_zero(float* __restrict__ p, long n) {
    long i = (long)blockIdx.x * blockDim.x + threadIdx.x;
    if (i < n) p[i] = 0.0f;
}

// ---------------------------------------------------------------------
// K4: COO SpMM via scatter-add.  One wave per nonzero; lane = float4 slice.
// ---------------------------------------------------------------------
__global__ void __launch_bounds__(256) k_spmm(
    const int*  __restrict__ rows,  // [T][nnz]
    const int*  __restrict__ cols,
    const float* __restrict__ vals,
    int nnz,
    const float* __restrict__ src, long src_stride,   // per-slice stride
    float* __restrict__ dst, long dst_stride)
{
    const int wave = threadIdx.x >> 5;
    const int lane = threadIdx.x & 31;
    const int t = blockIdx.y;
    const long e = (long)blockIdx.x * 8 + wave;
    if (e >= nnz) return;
    const long base = (long)t * nnz + e;
    const int   r = rows[base];
    const int   c = cols[base];
    const float v = vals[base];
    const float4 s = *(const float4*)(src + (long)t * src_stride + (long)c * H_ + lane * 4);
    float* d = dst + (long)t * dst_stride + (long)r * H_ + lane * 4;
    atomicAdd(d + 0, v * s.x);
    atomicAdd(d + 1, v * s.y);
    atomicAdd(d + 2, v * s.z);
    atomicAdd(d + 3, v * s.w);
}

// ---------------------------------------------------------------------
// K5: in-place relu (final layer activation)
// ---------------------------------------------------------------------
__global__ void k_relu(float* __restrict__ p, long n) {
    long i = (long)blockIdx.x * blockDim.x + threadIdx.x;
    if (i < n) p[i] = fmaxf(p[i], 0.0f);
}

// ---------------------------------------------------------------------
extern "C" void kernel_launch(void* const* d_in, const int* in_sizes, int n_in,
                              void* d_out, int out_size, void* d_ws, size_t ws_size,
                              hipStream_t stream) {
    const float* item_base = (const float*)d_in[0];
    const float* user_base = (const float*)d_in[1];
    const float* dy_table  = (const float*)d_in[2];
    const float* W_gate    = (const float*)d_in[3];
    const float* b_gate    = (const float*)d_in[4];
    const float* W_hgnn    = (const float*)d_in[5];
    const float* b_hgnn    = (const float*)d_in[6];
    const float* g_vals    = (const float*)d_in[7];
    const float* e_vals    = (const float*)d_in[8];
    const int*   rev_i     = (const int*)d_in[9];
    const int*   rev_dy    = (const int*)d_in[10];
    const int*   g_rows    = (const int*)d_in[11];
    const int*   g_cols    = (const int*)d_in[12];
    const int*   e_rows    = (const int*)d_in[13];
    const int*   e_cols    = (const int*)d_in[14];

    float* out  = (float*)d_out;
    // item-dynamic region of the output doubles as the x ping buffer
    float* xdyn = out + (long)(USER_TOT_ + ITEM_NUM_) * H_;   // [T*N][H]
    float* udst = out + (long)USER_NUM_ * H_;                 // user dynamic rows base
    float* ybuf = (float*)d_ws;                               // [T*N][H] (~82 MB)

    const long initN = (long)USER_TOT_ * H_ + (long)ITEM_NUM_ * H_;
    k_init<<<(unsigned)((initN + 255) / 256), 256, 0, stream>>>(user_base, item_base, out);

    dim3 gridGemm(625, T_);   // 1250 tiles of 16 rows, 2 waves/block
    k_gate<<<gridGemm, 64, 0, stream>>>(item_base, dy_table, W_gate, b_gate,
                                        rev_i, rev_dy, xdyn);

    const long dynN = (long)T_ * N_SUB * H_;
    dim3 gridG((NNZ_G_ + 7) / 8, T_);
    for (int l = 0; l < L_; ++l) {
        // y = act(x) W + b  (relu folded into the load for l>0)
        k_dense<<<gridGemm, 64, 0, stream>>>(xdyn, W_hgnn, b_hgnn, ybuf, l, l > 0 ? 1 : 0);
        // x = G @ y  (zero + scatter-add)
        k_zero<<<(unsigned)((dynN + 255) / 256), 256, 0, stream>>>(xdyn, dynN);
        k_spmm<<<gridG, 256, 0, stream>>>(g_rows, g_cols, g_vals, NNZ_G_,
                                          ybuf, (long)N_SUB * H_,
                                          xdyn, (long)N_SUB * H_);
    }
    // final activation in place -> item-dynamic output is now final
    k_relu<<<(unsigned)((dynN + 255) / 256), 256, 0, stream>>>(xdyn, dynN);

    // edges = E @ x  -> user dynamic rows (zeroed by k_init)
    dim3 gridE((NNZ_E_ + 7) / 8, T_);
    k_spmm<<<gridE, 256, 0, stream>>>(e_rows, e_cols, e_vals, NNZ_E_,
                                      xdyn, (long)N_SUB * H_,
                                      udst, (long)M_SUB * H_);
}